// LightGCN_27711128994140
// MI455X (gfx1250) — compile-verified
//
#include <hip/hip_runtime.h>

// LightGCN for MI455X (gfx1250, wave32).
// SpMM layers: L2-resident gather + global_atomic_add_f32 scatter (bandwidth/atomic bound).
// Scoring: V_WMMA_F32_16X16X4_F32, 16 pairs per wave, diagonal extraction.

typedef float v2f __attribute__((ext_vector_type(2)));
typedef float v8f __attribute__((ext_vector_type(8)));

#define N_TOTAL   500000LL
#define D_DIM     64LL
#define ND        (N_TOTAL * D_DIM)          // 32,000,000 floats
#define NNZ_EDGES 8000000LL
#define N_USERS_C 250000
#define BATCH_C   16384

// ---------------------------------------------------------------- vec helpers
__global__ void vec_copy4(float4* __restrict__ dst, const float4* __restrict__ src, long long n4) {
    long long i = (long long)blockIdx.x * blockDim.x + threadIdx.x;
    if (i < n4) dst[i] = src[i];
}

__global__ void vec_zero4(float4* __restrict__ dst, long long n4) {
    long long i = (long long)blockIdx.x * blockDim.x + threadIdx.x;
    if (i < n4) dst[i] = make_float4(0.f, 0.f, 0.f, 0.f);
}

__global__ void vec_accum4(float4* __restrict__ acc, const float4* __restrict__ add, long long n4) {
    long long i = (long long)blockIdx.x * blockDim.x + threadIdx.x;
    if (i < n4) {
        float4 a = acc[i], b = add[i];
        a.x += b.x; a.y += b.y; a.z += b.z; a.w += b.w;
        acc[i] = a;
    }
}

// ------------------------------------------------------------- SpMM scatter
// 16 threads per edge; each thread handles a float4 chunk of the 64-wide row.
// Edge streams (vals/rows/cols) loaded non-temporally so the L2 stays dedicated
// to the 128MB embedding/accumulator working set (fits in MI455X's 192MB L2).
__global__ void spmm_scatter(const float* __restrict__ vals,
                             const int*   __restrict__ rows2,   // int64 viewed as int pairs (low word)
                             const int*   __restrict__ cols2,
                             const float* __restrict__ src,
                             float*       __restrict__ dst,
                             long long nnz) {
    long long t = (long long)blockIdx.x * blockDim.x + threadIdx.x;
    long long e = t >> 4;
    if (e >= nnz) return;
    int q = (int)(t & 15);

    float v = __builtin_nontemporal_load(vals + e);
    int   r = __builtin_nontemporal_load(rows2 + 2 * e);
    int   c = __builtin_nontemporal_load(cols2 + 2 * e);

    const float4 g = *(const float4*)(src + (long long)c * D_DIM + q * 4);
    float* p = dst + (long long)r * D_DIM + q * 4;
    atomicAdd(p + 0, v * g.x);
    atomicAdd(p + 1, v * g.y);
    atomicAdd(p + 2, v * g.z);
    atomicAdd(p + 3, v * g.w);
}

// ------------------------------------------------------------- WMMA scoring
// One wave handles 16 (user,item) pairs. D = A(16x4) x B(4x16) accumulated over
// 16 K-slices of 4; diagonal D[i][i] = dot(user_i, item_i).
// f32 A layout: lanes 0-15 carry M=0..15 with K={k0,k0+1}; lanes 16-31 same M
// with K={k0+2,k0+3}. B mirrors this with N across lanes. So each lane loads an
// aligned float2 from its user row and item row at offset k0 + (lane>=16 ? 2:0).
__global__ void score_wmma(const float* __restrict__ acc,
                           const int*   __restrict__ x2,   // int64 pairs viewed as ints
                           float*       __restrict__ out) {
    const int lane = threadIdx.x & 31;
    const int wave = (int)blockIdx.x * (blockDim.x >> 5) + (threadIdx.x >> 5);
    const int base = wave * 16;
    const int m    = lane & 15;          // pair within wave (both lane halves share rows)
    const int p    = base + m;

    const int uidx = x2[4 * p + 0];                 // x[p][0] low word
    const int iidx = x2[4 * p + 2] + N_USERS_C;     // x[p][1] low word + N_USERS

    const float* urow = acc + (long long)uidx * D_DIM;
    const float* irow = acc + (long long)iidx * D_DIM;
    const int koff = (lane >= 16) ? 2 : 0;

    v8f c = {};
#pragma unroll
    for (int k0 = 0; k0 < 64; k0 += 4) {
        v2f a = *(const v2f*)(urow + k0 + koff);
        v2f b = *(const v2f*)(irow + k0 + koff);
        // 8 args: (neg_a, A, neg_b, B, c_mod, C, reuse_a, reuse_b)
        c = __builtin_amdgcn_wmma_f32_16x16x4_f32(false, a, false, b, (short)0, c,
                                                  false, false);
    }

    // Diagonal: score[i] at (VGPR i, lane i) for i<8; (VGPR i-8, lane i+16) for i>=8.
    const int sel = lane & 7;
    float v = c[0];
    v = (sel == 1) ? c[1] : v;
    v = (sel == 2) ? c[2] : v;
    v = (sel == 3) ? c[3] : v;
    v = (sel == 4) ? c[4] : v;
    v = (sel == 5) ? c[5] : v;
    v = (sel == 6) ? c[6] : v;
    v = (sel == 7) ? c[7] : v;
    v *= 0.0625f;  // (acc/4)·(acc/4) = acc·acc / 16

    if (lane < 8)        out[base + lane]      = v;   // i = lane
    else if (lane >= 24) out[base + lane - 16] = v;   // i = lane - 16

    if (blockIdx.x == 0 && threadIdx.x == 0) out[BATCH_C] = 0.0f;  // pruning loss
}

// -------------------------------------------------------------------- launch
extern "C" void kernel_launch(void* const* d_in, const int* in_sizes, int n_in,
                              void* d_out, int out_size, void* d_ws, size_t ws_size,
                              hipStream_t stream) {
    const float* emb      = (const float*)d_in[0];
    const float* adj_vals = (const float*)d_in[1];
    const int*   adj_rows = (const int*)d_in[2];   // int64 data, read low words
    const int*   adj_cols = (const int*)d_in[3];
    const int*   x        = (const int*)d_in[4];   // int64 data, read low words
    float*       out      = (float*)d_out;

    float* buf0 = (float*)d_ws;
    float* buf1 = buf0 + ND;
    float* acc  = buf1 + ND;

    const long long n4 = ND / 4;                       // 8,000,000 float4s
    const int vb = 256;
    const int vgrid = (int)((n4 + vb - 1) / vb);       // 31250 blocks
    const long long sthreads = NNZ_EDGES * 16;         // 128M threads
    const int sgrid = (int)((sthreads + 255) / 256);   // 500000 blocks

    // acc = emb
    vec_copy4<<<vgrid, vb, 0, stream>>>((float4*)acc, (const float4*)emb, n4);

    // Layer 1: buf0 = A @ emb ; acc += buf0
    vec_zero4<<<vgrid, vb, 0, stream>>>((float4*)buf0, n4);
    spmm_scatter<<<sgrid, 256, 0, stream>>>(adj_vals, adj_rows, adj_cols, emb, buf0, NNZ_EDGES);
    vec_accum4<<<vgrid, vb, 0, stream>>>((float4*)acc, (const float4*)buf0, n4);

    // Layer 2: buf1 = A @ buf0 ; acc += buf1
    vec_zero4<<<vgrid, vb, 0, stream>>>((float4*)buf1, n4);
    spmm_scatter<<<sgrid, 256, 0, stream>>>(adj_vals, adj_rows, adj_cols, buf0, buf1, NNZ_EDGES);
    vec_accum4<<<vgrid, vb, 0, stream>>>((float4*)acc, (const float4*)buf1, n4);

    // Layer 3: buf0 = A @ buf1 ; acc += buf0
    vec_zero4<<<vgrid, vb, 0, stream>>>((float4*)buf0, n4);
    spmm_scatter<<<sgrid, 256, 0, stream>>>(adj_vals, adj_rows, adj_cols, buf1, buf0, NNZ_EDGES);
    vec_accum4<<<vgrid, vb, 0, stream>>>((float4*)acc, (const float4*)buf0, n4);

    // Scoring: 16384 pairs, 16 per wave, 8 waves per block -> 128 blocks (all waves full).
    score_wmma<<<BATCH_C / 128, 256, 0, stream>>>(acc, x, out);
}